// SlidingWindow_80771154968643
// MI455X (gfx1250) — compile-verified
//
#include <hip/hip_runtime.h>

// Sliding-window gather: out[b,t,h,w,d] = (w < min(t+1,W)) ? src[b, max(0,t+1-W)+w, h, d] : 0
// Fixed shapes: B=1, T=2048, H=16, D=64, W=64. Pure data movement, write-BW bound
// (~1.07 GB f32 out -> ~46 us floor at 23.3 TB/s).
//
// CDNA5 strategy: block = (h, 16 consecutive t). The 16 windows share a 79-row
// span of the source; stage it ONCE in LDS via global_load_async_to_lds_b128,
// then emit each 16KB output tile as a contiguous LDS span with
// global_store_async_from_lds_b128 (no VGPR data movement). Boundary groups
// (t<64, zero-masked) use ds_load + nontemporal global_store_b128.

typedef __attribute__((ext_vector_type(4))) float f32x4;
typedef __attribute__((ext_vector_type(4))) int   v4i;   // b128 payload type

#if __has_builtin(__builtin_amdgcn_global_load_async_to_lds_b128) && \
    __has_builtin(__builtin_amdgcn_global_store_async_from_lds_b128)
#define HAS_ASYNC_LDS 1
#else
#define HAS_ASYNC_LDS 0
#endif

#define AS1 __attribute__((address_space(1)))
#define AS3 __attribute__((address_space(3)))

namespace {
constexpr int T = 2048;
constexpr int H = 16;
constexpr int D = 64;                       // floats per row (256 B)
constexpr int W = 64;
constexpr int TG = 16;                      // timesteps per block
constexpr int ROW_STRIDE = H * D;           // floats between consecutive t in src
constexpr int TILE = W * D;                 // 4096 floats per (t,h) output tile
constexpr int MAXROWS = W - 1 + TG;         // 79 source rows per group
} // namespace

__device__ __forceinline__ void wait_async_lds() {
#if __has_builtin(__builtin_amdgcn_s_wait_asynccnt)
  __builtin_amdgcn_s_wait_asynccnt(0);
#else
  asm volatile("s_wait_asynccnt 0x0" ::: "memory");
#endif
}

// Masked VGPR path: read window rows from LDS, zero-fill invalid w, NT-store.
__device__ __forceinline__ void store_masked(const float* lds, float* outT,
                                             int t0, int h, int rBase, int tid) {
  for (int tt = 0; tt < TG; ++tt) {
    const int t = t0 + tt;
    const int start = (t + 1 - W > 0) ? (t + 1 - W) : 0;
    const int vcnt  = (t + 1 < W) ? (t + 1) : W;
    const int lrow0 = start - rBase;
    f32x4* dst4 = reinterpret_cast<f32x4*>(outT + (size_t)(t * H + h) * TILE);
#pragma unroll
    for (int i = 0; i < 4; ++i) {
      const int idx4 = tid + i * 256;       // 0..1023
      const int w  = idx4 >> 4;
      const int d4 = idx4 & 15;
      f32x4 val = (f32x4)0.0f;
      if (w < vcnt)
        val = *reinterpret_cast<const f32x4*>(&lds[(size_t)(lrow0 + w) * D + d4 * 4]);
      __builtin_nontemporal_store(val, dst4 + idx4);
    }
  }
}

__global__ __launch_bounds__(256) void
sliding_window_async(const float* __restrict__ k, const float* __restrict__ v,
                     float* __restrict__ out) {
  const int bid = blockIdx.x;          // grp*H + h
  const int grp = bid >> 4;            // 0..127  (T/TG)
  const int h   = bid & (H - 1);
  const int isV = blockIdx.y;
  const int t0  = grp * TG;
  const int tid = (int)threadIdx.x;

  const float* __restrict__ src = isV ? v : k;
  float* __restrict__ outT = out + (size_t)isV * ((size_t)T * H * W * D);

  const int rBase  = (t0 > W - 1) ? (t0 - (W - 1)) : 0;  // first source row
  const int nRows  = (t0 + TG - 1) - rBase + 1;          // <= 79
  const int nChunks = nRows * (D / 4);                   // 16-byte chunks

  __shared__ float lds[(MAXROWS + 1) * D];               // 80 rows = 20 KB

  // ---- Load phase: stage the shared row window into LDS ----
#if HAS_ASYNC_LDS
  for (int c = tid; c < nChunks; c += 256) {
    const float* g = src + (size_t)(rBase + (c >> 4)) * ROW_STRIDE + h * D + (c & 15) * 4;
    __builtin_amdgcn_global_load_async_to_lds_b128(
        (AS1 v4i*)g, (AS3 v4i*)&lds[(size_t)c * 4], 0, 0);
  }
  wait_async_lds();
#else
  for (int c = tid; c < nChunks; c += 256) {
    const f32x4* g = reinterpret_cast<const f32x4*>(
        src + (size_t)(rBase + (c >> 4)) * ROW_STRIDE + h * D) + (c & 15);
    *reinterpret_cast<f32x4*>(&lds[(size_t)c * 4]) = *g;
  }
#endif
  __syncthreads();

  // ---- Store phase ----
#if HAS_ASYNC_LDS
  if (t0 >= W) {
    // Fully-valid groups: tile t is rows tt..tt+63 = a contiguous 16KB LDS span.
    // Pure LDS -> global async stores; data never touches VGPRs.
    for (int tt = 0; tt < TG; ++tt) {
      float* dst = outT + (size_t)((t0 + tt) * H + h) * TILE;
      const int ldsBase = tt * D;                        // float offset of row tt
#pragma unroll
      for (int i = 0; i < 4; ++i) {
        const int c = tid + i * 256;                     // 0..1023 chunk in tile
        __builtin_amdgcn_global_store_async_from_lds_b128(
            (AS1 v4i*)(dst + (size_t)c * 4),
            (AS3 v4i*)&lds[(size_t)(ldsBase + c * 4)], 0, 0);
      }
    }
    // s_endpgm's implicit wait-idle drains ASYNCcnt; LDS lives until WG end.
  } else {
    store_masked(lds, outT, t0, h, rBase, tid);          // boundary (t<64) groups
  }
#else
  store_masked(lds, outT, t0, h, rBase, tid);
#endif
}

extern "C" void kernel_launch(void* const* d_in, const int* in_sizes, int n_in,
                              void* d_out, int out_size, void* d_ws, size_t ws_size,
                              hipStream_t stream) {
  const float* k = (const float*)d_in[0];
  const float* v = (const float*)d_in[1];
  // d_in[2] = window_size (== 64, baked into compile-time constants).
  float* out = (float*)d_out;

  dim3 grid((T / TG) * H, 2);   // 2048 groups x {k_win, v_win}
  sliding_window_async<<<grid, 256, 0, stream>>>(k, v, out);
}